// _NonLocalLayer_6322191859973
// MI455X (gfx1250) — compile-verified
//
#include <hip/hip_runtime.h>

// MI455X / gfx1250 — wave32, WMMA f32 16x16x4.
//
// Reference collapses: sum(softmax(.,axis=2),axis=2)==1  =>  agg == V
//   => o = (Wout @ Wv) @ xf + bout   (precompute M = Wout@Wv)
// Then y = x + relu(o), followed by per-channel batchnorm.

typedef __attribute__((ext_vector_type(2))) float v2f;
typedef __attribute__((ext_vector_type(8))) float v8f;

#define CIN  256
#define FF   512
#define COUT 256
#define NB   8
#define NN   2048   // T*H*W = 8*16*16

// ---------------------------------------------------------------------------
// Kernel 1: M[256x256] = Wout[256x512] @ Wv[512x256], one wave per 16x16 tile.
// ---------------------------------------------------------------------------
__global__ __launch_bounds__(128) void k_fuse_weights(
    const float* __restrict__ Wout, const float* __restrict__ Wv,
    float* __restrict__ M) {
  const int wave = blockIdx.x * 4 + (threadIdx.x >> 5);   // 256 waves total
  const int lane = threadIdx.x & 31;
  const int h = lane >> 4, ln = lane & 15;
  const int m0 = (wave >> 4) * 16;   // output-channel tile
  const int c0 = (wave & 15) * 16;   // input-channel tile

  v8f acc = {};
  const float* arow = Wout + (size_t)(m0 + ln) * FF;
  for (int k0 = 0; k0 < FF; k0 += 4) {
    const int kk = k0 + 2 * h;                       // K pair for this lane-half
    v2f a = *(const v2f*)(arow + kk);                // A[m][kk], A[m][kk+1]
    v2f b;                                           // B[kk][c], B[kk+1][c]
    b.x = Wv[(size_t)kk * CIN + c0 + ln];
    b.y = Wv[(size_t)(kk + 1) * CIN + c0 + ln];
    acc = __builtin_amdgcn_wmma_f32_16x16x4_f32(false, a, false, b,
                                                (short)0, acc, false, false);
  }
#pragma unroll
  for (int r = 0; r < 8; ++r) {
    const int row = m0 + r + 8 * h;                  // C/D layout: VGPR r, half h
    M[(size_t)row * CIN + c0 + ln] = acc[r];
  }
}

// ---------------------------------------------------------------------------
// Kernel 2: per batch  o = M @ xf[b];  y = x + relu(o + bout)  -> d_out
// One wave computes a 16(rows) x 64(cols) strip: 4 accumulators share each A.
// ---------------------------------------------------------------------------
__global__ __launch_bounds__(128) void k_main_gemm(
    const float* __restrict__ x, const float* __restrict__ M,
    const float* __restrict__ bout, float* __restrict__ y) {
  const int wave = blockIdx.x * 4 + (threadIdx.x >> 5);   // 4096 waves
  const int lane = threadIdx.x & 31;
  const int h = lane >> 4, ln = lane & 15;

  const int b   = wave >> 9;          // 512 waves per batch
  const int rem = wave & 511;
  const int m0  = (rem >> 5) * 16;    // 16 row tiles
  const int n0  = (rem & 31) * 64;    // 32 col groups of 64

  const float* __restrict__ Xb   = x + (size_t)b * CIN * NN;
  const float* __restrict__ arow = M + (size_t)(m0 + ln) * CIN;

  v8f c0 = {}, c1 = {}, c2 = {}, c3 = {};
  for (int k0 = 0; k0 < CIN; k0 += 4) {
    const int kk = k0 + 2 * h;
    v2f a = *(const v2f*)(arow + kk);
    const float* xr0 = Xb + (size_t)kk * NN + n0 + ln;
    const float* xr1 = xr0 + NN;
    v2f b0, b1, b2, b3;
    b0.x = xr0[0];  b0.y = xr1[0];
    b1.x = xr0[16]; b1.y = xr1[16];
    b2.x = xr0[32]; b2.y = xr1[32];
    b3.x = xr0[48]; b3.y = xr1[48];
    c0 = __builtin_amdgcn_wmma_f32_16x16x4_f32(false, a, false, b0, (short)0, c0, false, false);
    c1 = __builtin_amdgcn_wmma_f32_16x16x4_f32(false, a, false, b1, (short)0, c1, false, false);
    c2 = __builtin_amdgcn_wmma_f32_16x16x4_f32(false, a, false, b2, (short)0, c2, false, false);
    c3 = __builtin_amdgcn_wmma_f32_16x16x4_f32(false, a, false, b3, (short)0, c3, false, false);
  }

#pragma unroll
  for (int r = 0; r < 8; ++r) {
    const int ch = m0 + r + 8 * h;
    const float bb = bout[ch];
    const size_t base = ((size_t)b * CIN + ch) * NN + n0 + ln;
    float o;
    o = c0[r] + bb; o = fmaxf(o, 0.f); y[base +  0] = x[base +  0] + o;
    o = c1[r] + bb; o = fmaxf(o, 0.f); y[base + 16] = x[base + 16] + o;
    o = c2[r] + bb; o = fmaxf(o, 0.f); y[base + 32] = x[base + 32] + o;
    o = c3[r] + bb; o = fmaxf(o, 0.f); y[base + 48] = x[base + 48] + o;
  }
}

// ---------------------------------------------------------------------------
// Kernel 3: per-channel mean / invstd over (B, N) -> stats[ch] = {mean, invstd}
// ---------------------------------------------------------------------------
__global__ __launch_bounds__(256) void k_bn_stats(
    const float* __restrict__ y, float* __restrict__ stats) {
  const int ch = blockIdx.x;
  float s = 0.f, sq = 0.f;
  for (int b = 0; b < NB; ++b) {
    const float* p = y + ((size_t)b * CIN + ch) * NN;
    for (int n = threadIdx.x; n < NN; n += 256) {
      const float v = p[n];
      s += v; sq += v * v;
    }
  }
  __shared__ float sh[256], shq[256];
  sh[threadIdx.x] = s; shq[threadIdx.x] = sq;
  __syncthreads();
  for (int off = 128; off > 0; off >>= 1) {
    if (threadIdx.x < off) {
      sh[threadIdx.x]  += sh[threadIdx.x + off];
      shq[threadIdx.x] += shq[threadIdx.x + off];
    }
    __syncthreads();
  }
  if (threadIdx.x == 0) {
    const float cnt  = (float)(NB * NN);
    const float mean = sh[0] / cnt;
    const float var  = shq[0] / cnt - mean * mean;
    stats[ch * 2 + 0] = mean;
    stats[ch * 2 + 1] = rsqrtf(var + 1e-5f);
  }
}

// ---------------------------------------------------------------------------
// Kernel 4: in-place normalize:  out = gamma*(y-mean)*invstd + beta
// ---------------------------------------------------------------------------
__global__ __launch_bounds__(256) void k_bn_apply(
    float* __restrict__ y, const float* __restrict__ stats,
    const float* __restrict__ gamma, const float* __restrict__ beta) {
  const size_t i = (size_t)blockIdx.x * 256 + threadIdx.x;
  const int ch = (int)((i / NN) & (CIN - 1));
  const float mean = stats[ch * 2 + 0];
  const float inv  = stats[ch * 2 + 1];
  y[i] = gamma[ch] * (y[i] - mean) * inv + beta[ch];
}

// ---------------------------------------------------------------------------
extern "C" void kernel_launch(void* const* d_in, const int* in_sizes, int n_in,
                              void* d_out, int out_size, void* d_ws, size_t ws_size,
                              hipStream_t stream) {
  (void)in_sizes; (void)n_in; (void)out_size; (void)ws_size;
  const float* x     = (const float*)d_in[0];
  // d_in[1] = Wk, d_in[2] = Wq : unused (softmax row-sum == 1 => attn is identity)
  const float* Wv    = (const float*)d_in[3];
  const float* Wout  = (const float*)d_in[4];
  const float* bout  = (const float*)d_in[5];
  const float* gamma = (const float*)d_in[6];
  const float* beta  = (const float*)d_in[7];

  float* out   = (float*)d_out;                             // y, then normalized in place
  float* M     = (float*)d_ws;                              // 256*256 fp32 = 256 KB
  float* stats = (float*)((char*)d_ws + CIN * COUT * 4);    // 256*2 fp32

  // 1) M = Wout @ Wv        (256 waves, one 16x16 tile each)
  k_fuse_weights<<<64, 128, 0, stream>>>(Wout, Wv, M);
  // 2) y = x + relu(M @ xf + bout)   (4096 waves, 16x64 strip each)
  k_main_gemm<<<1024, 128, 0, stream>>>(x, M, bout, out);
  // 3) per-channel stats
  k_bn_stats<<<COUT, 256, 0, stream>>>(out, stats);
  // 4) normalize in place
  k_bn_apply<<<(NB * CIN * NN) / 256, 256, 0, stream>>>(out, stats, gamma, beta);
}